// RetinaNetLoss_9698036154955
// MI455X (gfx1250) — compile-verified
//
#include <hip/hip_runtime.h>
#include <hip/hip_bf16.h>
#include <stdint.h>

// RetinaNet anchor<->gt matcher (torchvision Matcher(0.5, 0.4, allow_low_quality=True)).
// B=8 images, G=64 gt/image, A=120000 anchors. VALU-bound (~123M IoUs, ~15MB HBM).
// CDNA5 paths: global_load_async_to_lds_b128/b32 + s_wait_asynccnt staging,
// global_prefetch_b8, wave32 shfl_xor reductions, ds_max_u32 / global_atomic_max_u32.

#define NGT 64
#define TPB 256
#define HPG_BLKX 64   // anchor stripes per image in pass 1

// Identical IoU expression in both passes -> bit-identical q, so the
// low-quality-recovery equality test (q == highest_per_gt) matches reference.
__device__ __forceinline__ float iou_q(const float4 g, float areaG,
                                       const float4 a, float areaA) {
  float lx = fmaxf(g.x, a.x);
  float ly = fmaxf(g.y, a.y);
  float rx = fminf(g.z, a.z);
  float ry = fminf(g.w, a.w);
  float iw = fmaxf(rx - lx, 0.0f);
  float ih = fmaxf(ry - ly, 0.0f);
  float inter = iw * ih;
  return inter / (areaG + areaA - inter);   // union > 0 for valid boxes, IEEE div
}

__global__ void init_kernel(unsigned* __restrict__ p, int n) {
  int i = blockIdx.x * blockDim.x + threadIdx.x;
  if (i < n) p[i] = 0u;   // 0x0 == 0.0f; IoU >= 0 so uint-bit atomicMax is order-correct
}

// Pass 1: highest_per_gt[b][g] = max_a IoU. Each block walks one anchor stripe ONCE
// and keeps all 64 running maxima in registers (static indices via full unroll).
__global__ void __launch_bounds__(TPB)
hpg_kernel(const float4* __restrict__ gt, const float4* __restrict__ anchors,
           unsigned* __restrict__ highest_bits, int A) {
  const int b = blockIdx.y;

  __shared__ float4   sgt[NGT];
  __shared__ float    sarea[NGT];
  __shared__ unsigned smax[NGT];

  if (threadIdx.x < NGT) {
    uint32_t lds_gt = (uint32_t)(uintptr_t)(&sgt[threadIdx.x]);
    const float4* gsrc = gt + (size_t)b * NGT + threadIdx.x;
    asm volatile("global_load_async_to_lds_b128 %0, %1, off"
                 :: "v"(lds_gt), "v"(gsrc) : "memory");
  }
  asm volatile("s_wait_asynccnt 0" ::: "memory");
  __syncthreads();
  if (threadIdx.x < NGT) {
    float4 g4 = sgt[threadIdx.x];
    sarea[threadIdx.x] = (g4.z - g4.x) * (g4.w - g4.y);
    smax[threadIdx.x]  = 0u;
  }
  __syncthreads();

  float m[NGT];
  #pragma unroll
  for (int g = 0; g < NGT; ++g) m[g] = 0.0f;

  const int stride = gridDim.x * TPB;
  for (int a = blockIdx.x * TPB + threadIdx.x; a < A; a += stride) {
    __builtin_prefetch(&anchors[a + stride], 0, 0);   // global_prefetch_b8
    float4 an = anchors[a];                            // global_load_b128
    float areaA = (an.z - an.x) * (an.w - an.y);
    #pragma unroll
    for (int g = 0; g < NGT; ++g) {
      m[g] = fmaxf(m[g], iou_q(sgt[g], sarea[g], an, areaA));
    }
  }

  // wave32 butterfly max per gt, then LDS atomic max (ds_max_u32 on IoU bits)
  const int lane = threadIdx.x & 31;
  #pragma unroll
  for (int g = 0; g < NGT; ++g) {
    float v = m[g];
    #pragma unroll
    for (int s = 16; s > 0; s >>= 1) v = fmaxf(v, __shfl_xor(v, s, 32));
    if (lane == 0) atomicMax(&smax[g], __float_as_uint(v));
  }
  __syncthreads();

  if (threadIdx.x < NGT)   // one global_atomic_max_u32 per gt per block
    atomicMax(&highest_bits[b * NGT + threadIdx.x], smax[threadIdx.x]);
}

// Pass 2: per-anchor max/argmax over g, thresholds, low-quality recovery.
__global__ void __launch_bounds__(TPB)
match_kernel(const float4* __restrict__ gt, const float4* __restrict__ anchors,
             const unsigned* __restrict__ highest_bits,
             long long* __restrict__ out_idx, float* __restrict__ out_val, int A) {
  const int b = blockIdx.y;

  __shared__ float4 sgt[NGT];
  __shared__ float2 smeta[NGT];   // {areaG, highest_per_gt}

  if (threadIdx.x < NGT) {
    uint32_t lds_gt = (uint32_t)(uintptr_t)(&sgt[threadIdx.x]);
    const float4* gsrc = gt + (size_t)b * NGT + threadIdx.x;
    asm volatile("global_load_async_to_lds_b128 %0, %1, off"
                 :: "v"(lds_gt), "v"(gsrc) : "memory");
    uint32_t lds_hi = (uint32_t)(uintptr_t)(&smeta[threadIdx.x].y);
    const unsigned* hsrc = highest_bits + b * NGT + threadIdx.x;
    asm volatile("global_load_async_to_lds_b32 %0, %1, off"
                 :: "v"(lds_hi), "v"(hsrc) : "memory");
  }
  asm volatile("s_wait_asynccnt 0" ::: "memory");
  __syncthreads();
  if (threadIdx.x < NGT) {
    float4 g4 = sgt[threadIdx.x];
    smeta[threadIdx.x].x = (g4.z - g4.x) * (g4.w - g4.y);
  }
  __syncthreads();

  const int a = blockIdx.x * TPB + threadIdx.x;
  if (a >= A) return;

  float4 an = anchors[a];                              // global_load_b128
  const float areaA = (an.z - an.x) * (an.w - an.y);

  float bestQ = -1.0f;   // first q (>= 0) always wins
  int   bestG = 0;
  int   anyHigh = 0;

  #pragma unroll
  for (int g = 0; g < NGT; ++g) {
    float2 mt = smeta[g];                              // ds_load_b64 broadcast
    float q = iou_q(sgt[g], mt.x, an, areaA);          // ds_load_b128 broadcast
    if (q > bestQ) { bestQ = q; bestG = g; }           // strict '>' = first-tie argmax
    anyHigh |= (q == mt.y);                            // exact-equality recovery
  }

  long long mi = (long long)bestG;                     // pred_update keeps all_matches
  if (!anyHigh) {
    if (bestQ < 0.4f)      mi = -1;                    // BELOW_LOW_QUALITY
    else if (bestQ < 0.5f) mi = -2;                    // BETWEEN_THRESHOLDS
  }

  const size_t o = (size_t)b * A + a;
  out_idx[o] = mi;
  out_val[o] = bestQ;
}

extern "C" void kernel_launch(void* const* d_in, const int* in_sizes, int n_in,
                              void* d_out, int out_size, void* d_ws, size_t ws_size,
                              hipStream_t stream) {
  const float4* gt      = (const float4*)d_in[0];   // [B, 64, 4] f32
  const float4* anchors = (const float4*)d_in[1];   // [A, 4] f32
  const int B = in_sizes[0] / (NGT * 4);
  const int A = in_sizes[1] / 4;

  unsigned* highest_bits = (unsigned*)d_ws;         // [B*64] IoU max as uint bits

  // d_out = concat(matched_idxs int64 [B,A], matched_vals f32 [B,A]) flat bytes.
  long long* out_idx = (long long*)d_out;
  float*     out_val = (float*)(out_idx + (size_t)B * A);

  init_kernel<<<dim3((B * NGT + TPB - 1) / TPB), TPB, 0, stream>>>(highest_bits, B * NGT);
  hpg_kernel<<<dim3(HPG_BLKX, B), TPB, 0, stream>>>(gt, anchors, highest_bits, A);
  match_kernel<<<dim3((A + TPB - 1) / TPB, B), TPB, 0, stream>>>(
      gt, anchors, highest_bits, out_idx, out_val, A);
}